// ARMA_30374008717356
// MI455X (gfx1250) — compile-verified
//
#include <hip/hip_runtime.h>

typedef float v2f __attribute__((ext_vector_type(2)));
typedef float v8f __attribute__((ext_vector_type(8)));

// ---------------------------------------------------------------- utilities
__global__ void zero_f32(float* __restrict__ p, int n) {
  int i = blockIdx.x * blockDim.x + threadIdx.x;
  if (i < n) p[i] = 0.f;
}

__global__ void relu_inplace(float* __restrict__ p, int n) {
  int i = blockIdx.x * blockDim.x + threadIdx.x;
  if (i < n) p[i] = fmaxf(p[i], 0.f);
}

// ------------------------------------------------------- degree + gcn norm
__global__ void deg_accum(const int* __restrict__ dst, const float* __restrict__ w,
                          float* __restrict__ deg, int E) {
  int e = blockIdx.x * blockDim.x + threadIdx.x;
  if (e < E) atomicAdd(deg + dst[e], w[e]);
}

__global__ void norm_compute(const int* __restrict__ src, const int* __restrict__ dst,
                             const float* __restrict__ w, const float* __restrict__ deg,
                             float* __restrict__ norm, int E) {
  int e = blockIdx.x * blockDim.x + threadIdx.x;
  if (e >= E) return;
  float ds = deg[src[e]];
  float dd = deg[dst[e]];
  float is = ds > 0.f ? rsqrtf(ds) : 0.f;
  float id = dd > 0.f ? rsqrtf(dd) : 0.f;
  norm[e] = is * w[e] * id;
}

// ---------------------------------------------------------- WMMA f32 GEMM
// C[N x F] = A[N x K] * B[K x F] (+ bias), row-major. One wave per 16x16 tile.
// V_WMMA_F32_16X16X4_F32; OOB lanes load CLAMPED addresses (A row i only
// affects C row i, B col j only C col j -> garbage never stored), so the
// K loop has no EXEC manipulation and fully unrolls into a load clause
// followed by back-to-back WMMAs.
template <int K>
__global__ void gemm_wmma(const float* __restrict__ A, const float* __restrict__ B,
                          const float* __restrict__ bias, float* __restrict__ C,
                          int N, int F, int rowTiles, int colTiles) {
  const int lane = threadIdx.x & 31;
  const int wave = threadIdx.x >> 5;
  const int tile = blockIdx.x * (blockDim.x >> 5) + wave;
  if (tile >= rowTiles * colTiles) return;   // wave-uniform exit

  const int rt = tile / colTiles;
  const int ct = tile - rt * colTiles;
  const int row0 = rt << 4;
  const int col0 = ct << 4;
  const int half = lane >> 4;     // 0 for lanes 0-15, 1 for lanes 16-31
  const int lm   = lane & 15;

  const bool cOK = (col0 + lm) < F;
  const int  arow = min(row0 + lm, N - 1);   // clamped: safe to load
  const int  bcol = min(col0 + lm, F - 1);   // clamped: safe to load

  const float* Arow = A + (size_t)arow * K + 2 * half;
  const float* Bcol = B + bcol;

  v2f afrag[K / 4];
  v2f bfrag[K / 4];
#pragma unroll
  for (int i = 0; i < K / 4; ++i) {
    const int ka = 4 * i + 2 * half;         // even -> 8B-aligned float2
    afrag[i] = *reinterpret_cast<const v2f*>(Arow + 4 * i);
    bfrag[i].x = Bcol[(size_t)ka * F];
    bfrag[i].y = Bcol[(size_t)(ka + 1) * F];
  }

  v8f acc = {};
#pragma unroll
  for (int i = 0; i < K / 4; ++i) {
    acc = __builtin_amdgcn_wmma_f32_16x16x4_f32(
        false, afrag[i], false, bfrag[i], (short)0, acc, false, false);
  }

  const float bv = (bias != nullptr) ? bias[bcol] : 0.f;
#pragma unroll
  for (int r = 0; r < 8; ++r) {
    const int row = row0 + r + 8 * half;     // C layout: VGPR r -> M = r + 8*half
    if (row < N && cOK) C[(size_t)row * F + col0 + lm] = acc[r] + bv;
  }
}

// Generic-K fallback (same clamping scheme, runtime loop).
__global__ void gemm_wmma_gen(const float* __restrict__ A, const float* __restrict__ B,
                              const float* __restrict__ bias, float* __restrict__ C,
                              int N, int K, int F, int rowTiles, int colTiles) {
  const int lane = threadIdx.x & 31;
  const int wave = threadIdx.x >> 5;
  const int tile = blockIdx.x * (blockDim.x >> 5) + wave;
  if (tile >= rowTiles * colTiles) return;
  const int rt = tile / colTiles;
  const int ct = tile - rt * colTiles;
  const int row0 = rt << 4, col0 = ct << 4;
  const int half = lane >> 4, lm = lane & 15;
  const bool cOK = (col0 + lm) < F;
  const int arow = min(row0 + lm, N - 1);
  const int bcol = min(col0 + lm, F - 1);
  v8f acc = {};
  for (int kb = 0; kb < K; kb += 4) {
    const int ka = kb + 2 * half;
    v2f a = *reinterpret_cast<const v2f*>(A + (size_t)arow * K + ka);
    v2f b;
    b.x = B[(size_t)ka * F + bcol];
    b.y = B[(size_t)(ka + 1) * F + bcol];
    acc = __builtin_amdgcn_wmma_f32_16x16x4_f32(false, a, false, b, (short)0, acc,
                                                false, false);
  }
  const float bv = (bias != nullptr) ? bias[bcol] : 0.f;
#pragma unroll
  for (int r = 0; r < 8; ++r) {
    const int row = row0 + r + 8 * half;
    if (row < N && cOK) C[(size_t)row * F + col0 + lm] = acc[r] + bv;
  }
}

// --------------------------------------------------- edge gather + scatter
// One thread per (edge, group of 4 features): float4 gather, 4 f32 atomics.
__global__ void edge_scatter(const float* __restrict__ h, const float* __restrict__ norm,
                             const int* __restrict__ src, const int* __restrict__ dst,
                             float* __restrict__ agg, int E, int F, int G) {
  long long tid = (long long)blockIdx.x * blockDim.x + threadIdx.x;
  if (tid >= (long long)E * G) return;
  int e = (int)(tid / G);
  int g = (int)(tid - (long long)e * G);
  int s = src[e];
  int d = dst[e];
  float w = norm[e];
  const float4 v = *reinterpret_cast<const float4*>(h + (size_t)s * F + (size_t)g * 4);
  float* base = agg + (size_t)d * F + (size_t)g * 4;
  atomicAdd(base + 0, v.x * w);
  atomicAdd(base + 1, v.y * w);
  atomicAdd(base + 2, v.z * w);
  atomicAdd(base + 3, v.w * w);
}

// ------------------------------------------- fused ReLU + log_softmax
template <int C>
__global__ void relu_logsoftmax_t(float* __restrict__ y, int N) {
  int n = blockIdx.x * blockDim.x + threadIdx.x;
  if (n >= N) return;
  float v[C];
  float m = -3.4e38f;
  float* row = y + (size_t)n * C;
#pragma unroll
  for (int c = 0; c < C; ++c) {
    float t = fmaxf(row[c], 0.f);   // ReLU of conv-2 output
    v[c] = t;
    m = fmaxf(m, t);
  }
  float s = 0.f;
#pragma unroll
  for (int c = 0; c < C; ++c) s += __expf(v[c] - m);
  float ls = m + __logf(s);
#pragma unroll
  for (int c = 0; c < C; ++c) row[c] = v[c] - ls;
}

__global__ void relu_logsoftmax_gen(float* __restrict__ y, int N, int C) {
  int n = blockIdx.x * blockDim.x + threadIdx.x;
  if (n >= N) return;
  float* row = y + (size_t)n * C;
  float m = -3.4e38f;
  for (int c = 0; c < C; ++c) m = fmaxf(m, fmaxf(row[c], 0.f));
  float s = 0.f;
  for (int c = 0; c < C; ++c) s += __expf(fmaxf(row[c], 0.f) - m);
  float ls = m + __logf(s);
  for (int c = 0; c < C; ++c) row[c] = fmaxf(row[c], 0.f) - ls;
}

// ------------------------------------------------------------------ driver
static inline int ceil_div(long long a, int b) { return (int)((a + b - 1) / b); }

static void launch_gemm(const float* A, const float* B, const float* bias, float* C,
                        int N, int K, int F, hipStream_t stream) {
  const int TPB = 256;
  int rowT = ceil_div(N, 16), colT = ceil_div(F, 16);
  int blocks = ceil_div((long long)rowT * colT, TPB / 32);
  if (K == 64) {
    gemm_wmma<64><<<blocks, TPB, 0, stream>>>(A, B, bias, C, N, F, rowT, colT);
  } else if (K == 48) {
    gemm_wmma<48><<<blocks, TPB, 0, stream>>>(A, B, bias, C, N, F, rowT, colT);
  } else {
    gemm_wmma_gen<<<blocks, TPB, 0, stream>>>(A, B, bias, C, N, K, F, rowT, colT);
  }
}

extern "C" void kernel_launch(void* const* d_in, const int* in_sizes, int n_in,
                              void* d_out, int out_size, void* d_ws, size_t ws_size,
                              hipStream_t stream) {
  const float* x  = (const float*)d_in[0];
  const int*   ei = (const int*)d_in[1];
  const float* ew = (const float*)d_in[2];
  const float* W1 = (const float*)d_in[3];
  const float* V1 = (const float*)d_in[4];
  const float* b1 = (const float*)d_in[5];
  const float* W2 = (const float*)d_in[6];
  const float* V2 = (const float*)d_in[7];
  const float* b2 = (const float*)d_in[8];
  float* out = (float*)d_out;

  const int H   = in_sizes[5];             // 48
  const int Cn  = in_sizes[8];             // 40
  const int Fin = in_sizes[3] / H;         // 64
  const int N   = in_sizes[0] / Fin;       // 100000
  const int E   = in_sizes[2];             // 1600000
  const int* src = ei;
  const int* dst = ei + E;

  // workspace layout (floats), 256-float aligned offsets
  float* ws = (float*)d_ws;
  size_t off = 0;
  float* deg  = ws + off; off += ((size_t)N + 255) & ~(size_t)255;
  float* norm = ws + off; off += ((size_t)E + 255) & ~(size_t)255;
  float* hbuf = ws + off; off += ((size_t)N * H + 255) & ~(size_t)255;  // reused for layer 2
  float* z1   = ws + off; off += ((size_t)N * H + 255) & ~(size_t)255;

  const int TPB = 256;

  // ---- gcn_norm
  zero_f32<<<ceil_div(N, TPB), TPB, 0, stream>>>(deg, N);
  deg_accum<<<ceil_div(E, TPB), TPB, 0, stream>>>(dst, ew, deg, E);
  norm_compute<<<ceil_div(E, TPB), TPB, 0, stream>>>(src, dst, ew, deg, norm, E);

  // ---- layer 1: h = x@W1 ; z1 = x@V1 + b1 ; z1 += scatter(h[src]*norm) ; relu
  launch_gemm(x, W1, nullptr, hbuf, N, Fin, H, stream);
  launch_gemm(x, V1, b1,      z1,   N, Fin, H, stream);
  {
    int G = H / 4;
    edge_scatter<<<ceil_div((long long)E * G, TPB), TPB, 0, stream>>>(
        hbuf, norm, src, dst, z1, E, H, G);
    relu_inplace<<<ceil_div((long long)N * H, TPB), TPB, 0, stream>>>(z1, N * H);
  }

  // ---- layer 2: h = z1@W2 ; out = z1@V2 + b2 ; out += scatter ; relu+log_softmax
  launch_gemm(z1, W2, nullptr, hbuf, N, H, Cn, stream);
  launch_gemm(z1, V2, b2,      out,  N, H, Cn, stream);
  {
    int G = Cn / 4;
    edge_scatter<<<ceil_div((long long)E * G, TPB), TPB, 0, stream>>>(
        hbuf, norm, src, dst, out, E, Cn, G);
    if (Cn == 40) {
      relu_logsoftmax_t<40><<<ceil_div(N, TPB), TPB, 0, stream>>>(out, N);
    } else {
      relu_logsoftmax_gen<<<ceil_div(N, TPB), TPB, 0, stream>>>(out, N, Cn);
    }
  }
}